// RSSD_1C_42073499631627
// MI455X (gfx1250) — compile-verified
//
#include <hip/hip_runtime.h>
#include <math.h>

typedef float v2f __attribute__((ext_vector_type(2)));
typedef float v8f __attribute__((ext_vector_type(8)));

#define B_IMG  32
#define A_N    24564
#define C_N    21
#define NC     20          // foreground classes
#define TOPK   200
#define CONF_T 0.01f
#define NMS_T  0.45f

// ---------------- Stage 0: zero the whole output (background rows stay 0) ---
__global__ void k_zero(float* __restrict__ out, int n) {
  int i = blockIdx.x * blockDim.x + threadIdx.x;
  if (i < n) out[i] = 0.0f;
}

// ---------------- Stage 1: softmax; WMMA computes per-anchor exp-sums -------
// Each wave handles 16 anchors. Lane L (L&15 = anchor m, L>>4 = h) holds
// classes {4k+2h, 4k+2h+1} for k=0..5 (24 slots; c>=21 padded with 0).
// D = ones(16x4) x E(4x16) accumulated over 6 K-chunks => D[:,m] = sum_c e[m][c],
// and d[0] in every lane is the sum for anchor (lane&15).
__global__ void __launch_bounds__(256)
k_softmax(const float* __restrict__ cls, float* __restrict__ scores) {
  const int lane = threadIdx.x & 31;
  const int wid  = threadIdx.x >> 5;
  const int gw   = blockIdx.x * (blockDim.x >> 5) + wid;   // global wave id
  const int tilesPer = (A_N + 15) / 16;                    // 1536
  const int b = gw / tilesPer;
  const int t = gw % tilesPer;
  if (b >= B_IMG) return;                                  // wave-uniform

  const int m = lane & 15;
  const int h = lane >> 4;
  const int a = t * 16 + m;
  const bool valid = (a < A_N);
  const int ac = valid ? a : 0;                            // clamp address
  const float* base = cls + ((long)b * A_N + ac) * C_N;

  // load this lane's 12 class logits, track max over valid classes
  float x[12];
  float mx = -1e30f;
  #pragma unroll
  for (int kk = 0; kk < 6; ++kk) {
    #pragma unroll
    for (int j = 0; j < 2; ++j) {
      int c = 4 * kk + 2 * h + j;
      float v = (c < C_N) ? base[c] : -1e30f;
      x[kk * 2 + j] = v;
      mx = fmaxf(mx, v);
    }
  }
  // combine the two half-wave partners that share an anchor
  mx = fmaxf(mx, __shfl_xor(mx, 16, 32));

  float e[12];
  #pragma unroll
  for (int kk = 0; kk < 6; ++kk) {
    #pragma unroll
    for (int j = 0; j < 2; ++j) {
      int c = 4 * kk + 2 * h + j;
      e[kk * 2 + j] = (c < C_N) ? expf(x[kk * 2 + j] - mx) : 0.0f;
    }
  }

  // WMMA ones-matmul: exact f32 column sums over 6 chained accumulations
  v2f onesA = {1.0f, 1.0f};
  v8f acc = {};
  #pragma unroll
  for (int kk = 0; kk < 6; ++kk) {
    v2f bv = {e[kk * 2], e[kk * 2 + 1]};
    acc = __builtin_amdgcn_wmma_f32_16x16x4_f32(
        /*neg_a=*/false, onesA, /*neg_b=*/false, bv,
        /*c_mod=*/(short)0, acc, /*reuse_a=*/false, /*reuse_b=*/false);
  }
  const float inv = 1.0f / acc[0];

  if (valid) {
    #pragma unroll
    for (int kk = 0; kk < 6; ++kk) {
      #pragma unroll
      for (int j = 0; j < 2; ++j) {
        int c = 4 * kk + 2 * h + j;
        if (c >= 1 && c < C_N)
          scores[((long)(b * NC + (c - 1))) * A_N + a] = e[kk * 2 + j] * inv;
      }
    }
  }
}

// ---------------- Stage 2: SSD box decode ----------------------------------
__global__ void k_decode(const float* __restrict__ loc,
                         const float* __restrict__ anc,
                         float* __restrict__ boxes) {
  long i = (long)blockIdx.x * blockDim.x + threadIdx.x;
  const long n = (long)B_IMG * A_N;
  if (i >= n) return;
  const int a = (int)(i % A_N);
  const float* l = loc + i * 4;
  const float* p = anc + (long)a * 4;
  float cx = p[0] + l[0] * 0.1f * p[2];
  float cy = p[1] + l[1] * 0.1f * p[3];
  float w  = p[2] * expf(l[2] * 0.2f);
  float h  = p[3] * expf(l[3] * 0.2f);
  float* o = boxes + i * 4;
  o[0] = cx - 0.5f * w; o[1] = cy - 0.5f * h;
  o[2] = cx + 0.5f * w; o[3] = cy + 0.5f * h;
}

// ---------------- Stage 3: per-(image,class) top-200 + greedy NMS ----------
__global__ void __launch_bounds__(256)
k_topk_nms(const float* __restrict__ scores, const float* __restrict__ boxes,
           float* __restrict__ out) {
  __shared__ float s_sc[A_N];           // 98.3 KB score stage (320KB WGP LDS)
  __shared__ float red_v[8];
  __shared__ int   red_i[8];
  __shared__ float top_v[TOPK];
  __shared__ int   top_i[TOPK];
  __shared__ float tb[TOPK][4];
  __shared__ float tarea[TOPK];
  __shared__ int   keepA[TOPK];

  const int tid  = threadIdx.x;
  const int b    = blockIdx.x / NC;
  const int cl   = blockIdx.x % NC;     // foreground class index (c = cl+1)
  const float* sc = scores + ((long)(b * NC + cl)) * A_N;

  for (int j = tid; j < A_N; j += 256) {
    float v = sc[j];
    s_sc[j] = (v > CONF_T) ? v : 0.0f;  // scores_m = where(s>thr, s, 0)
  }
  __syncthreads();

  const int lane = tid & 31, wid = tid >> 5;
  for (int k = 0; k < TOPK; ++k) {
    // arg-max with lax.top_k tie-break: highest value, lowest index
    float bv = -2.0f; int bi = 0;
    for (int j = tid; j < A_N; j += 256) {
      float v = s_sc[j];
      if (v > bv) { bv = v; bi = j; }   // ascending scan keeps lowest index
    }
    #pragma unroll
    for (int off = 16; off > 0; off >>= 1) {
      float ov = __shfl_down(bv, off, 32);
      int   oi = __shfl_down(bi, off, 32);
      if (ov > bv || (ov == bv && oi < bi)) { bv = ov; bi = oi; }
    }
    if (lane == 0) { red_v[wid] = bv; red_i[wid] = bi; }
    __syncthreads();
    if (tid == 0) {
      float fv = red_v[0]; int fi = red_i[0];
      #pragma unroll
      for (int w = 1; w < 8; ++w) {
        float ov = red_v[w]; int oi = red_i[w];
        if (ov > fv || (ov == fv && oi < fi)) { fv = ov; fi = oi; }
      }
      top_v[k] = fv; top_i[k] = fi;
      s_sc[fi] = -1.0f;                 // remove from further selection
    }
    __syncthreads();
  }

  if (tid < TOPK) {
    const float* bp = boxes + ((long)b * A_N + top_i[tid]) * 4;
    float x1 = bp[0], y1 = bp[1], x2 = bp[2], y2 = bp[3];
    tb[tid][0] = x1; tb[tid][1] = y1; tb[tid][2] = x2; tb[tid][3] = y2;
    tarea[tid] = fmaxf(x2 - x1, 0.0f) * fmaxf(y2 - y1, 0.0f);
    keepA[tid] = 1;
  }
  __syncthreads();

  // greedy NMS: i suppresses j>i iff keep[i] at iteration i (matches fori_loop)
  for (int i = 0; i < TOPK; ++i) {
    if (keepA[i]) {
      if (tid < TOPK && tid > i && keepA[tid]) {
        float lx = fmaxf(tb[i][0], tb[tid][0]);
        float ly = fmaxf(tb[i][1], tb[tid][1]);
        float rx = fminf(tb[i][2], tb[tid][2]);
        float ry = fminf(tb[i][3], tb[tid][3]);
        float inter = fmaxf(rx - lx, 0.0f) * fmaxf(ry - ly, 0.0f);
        float iou = inter / (tarea[i] + tarea[tid] - inter + 1e-9f);
        if (iou > NMS_T) keepA[tid] = 0;
      }
    }
    __syncthreads();
  }

  if (tid < TOPK) {
    float kf = (keepA[tid] && top_v[tid] > 0.0f) ? 1.0f : 0.0f;
    long o = (((long)b * C_N + (cl + 1)) * TOPK + tid) * 5;
    out[o + 0] = top_v[tid] * kf;
    out[o + 1] = tb[tid][0] * kf;
    out[o + 2] = tb[tid][1] * kf;
    out[o + 3] = tb[tid][2] * kf;
    out[o + 4] = tb[tid][3] * kf;
  }
}

// ---------------------------------------------------------------------------
extern "C" void kernel_launch(void* const* d_in, const int* in_sizes, int n_in,
                              void* d_out, int out_size, void* d_ws, size_t ws_size,
                              hipStream_t stream) {
  const float* cls = (const float*)d_in[0];   // [32, 24564, 21]
  const float* loc = (const float*)d_in[1];   // [32, 24564, 4]
  const float* anc = (const float*)d_in[2];   // [24564, 4]
  float* out = (float*)d_out;                 // [32, 21, 200, 5]

  char* ws = (char*)d_ws;
  float* scores = (float*)ws;                                   // 62.9 MB
  float* boxes  = (float*)(ws + (size_t)B_IMG * NC * A_N * 4);  // 12.6 MB

  k_zero<<<(out_size + 255) / 256, 256, 0, stream>>>(out, out_size);

  {
    const int tiles  = (A_N + 15) / 16;          // 1536
    const int waves  = B_IMG * tiles;            // 49152
    const int blocks = (waves + 7) / 8;          // 6144 (8 waves/block)
    k_softmax<<<blocks, 256, 0, stream>>>(cls, scores);
  }
  {
    const long n = (long)B_IMG * A_N;
    k_decode<<<(int)((n + 255) / 256), 256, 0, stream>>>(loc, anc, boxes);
  }
  k_topk_nms<<<B_IMG * NC, 256, 0, stream>>>(scores, boxes, out);
}